// MeanPool_8297876815923
// MI455X (gfx1250) — compile-verified
//
#include <hip/hip_runtime.h>

// ---------------------------------------------------------------------------
// Segment mean-pool (global_mean_pool) for MI455X / gfx1250.
//
// Roofline: 1.024 GB streamed once at 23.3 TB/s -> ~44 us floor; pure
// bandwidth problem. batch[] is sorted, so each segment is a contiguous row
// range: one workgroup per segment, boundaries via binary search (no
// atomics, deterministic). Row folding runs on the matrix pipe:
// V_WMMA_F32_16X16X4_F32 with an all-ones A (layout-invariant) sums 4 rows
// per issue in full fp32. Bulk loop is branch-free with immediate-offset
// loads from a single striding base pointer; <=15-row tail is a uniform
// scalar loop (merged only into the half-0 lanes that store).
// ---------------------------------------------------------------------------

typedef float v2f __attribute__((ext_vector_type(2)));
typedef float v8f __attribute__((ext_vector_type(8)));

#define FDIM 256   // feature dimension (fixed by the reference)

__device__ __forceinline__ int lower_bound_i64(const long long* __restrict__ b,
                                               int n, long long key) {
  int lo = 0, hi = n;
  while (lo < hi) {
    int mid = (lo + hi) >> 1;
    if (b[mid] < key) lo = mid + 1; else hi = mid;
  }
  return lo;
}

__global__ __launch_bounds__(512)
void MeanPoolWmmaKernel(const float* __restrict__ x,
                        const long long* __restrict__ batch,
                        float* __restrict__ out, int n) {
  const int g    = blockIdx.x;        // segment id, 0..1023
  const int t    = threadIdx.x;       // 0..511 -> 16 waves (wave32)
  const int wave = t >> 5;            // 16-feature tile per wave
  const int lane = t & 31;
  const int col  = lane & 15;         // column within tile (N = lane % 16)
  const int half = lane >> 4;         // half-wave selects which K rows we feed
  const int f    = (wave << 4) + col; // absolute feature 0..255

  // Contiguous row range of this segment (batch is sorted).
  const int s   = lower_bound_i64(batch, n, (long long)g);
  const int e   = lower_bound_i64(batch, n, (long long)g + 1);
  const int cnt = e - s;

  // A = all-ones 16x4 fp32 matrix: layout-invariant (every lane, every VGPR
  // holds 1.0, and ones*B sums over all K regardless of K-slot permutation).
  v2f a; a.x = 1.0f; a.y = 1.0f;

  // Four independent accumulators: 16 rows in flight per iteration, no
  // serial WMMA->WMMA dependency inside the loop body.
  v8f c0 = {}, c1 = {}, c2 = {}, c3 = {};

  const float* __restrict__ xf = x + f;   // this lane's feature column

  // ---- Bulk: multiple-of-16 rows, fully unconditional loads --------------
  const int bulk = cnt & ~15;
  // Lane base: half-wave h starts at row s + 2h; per 4-row group j it loads
  // rows 4j + 2h and 4j + 2h + 1 -> float offsets 1024*j and 1024*j + 256
  // (byte offsets 0..13312: all fold into the 24-bit instruction offset).
  const float* __restrict__ p = xf + (size_t)(s + 2 * half) * FDIM;

  for (int i = 0; i < bulk; i += 16) {
    if (i + 64 < cnt) {                       // uniform guard, scalar branch
      __builtin_prefetch(p + 64 * FDIM, 0, 1);  // global_prefetch_b8
    }
    v2f b0 = { p[0],    p[256]  };
    v2f b1 = { p[1024], p[1280] };
    v2f b2 = { p[2048], p[2304] };
    v2f b3 = { p[3072], p[3328] };
    c0 = __builtin_amdgcn_wmma_f32_16x16x4_f32(false, a, false, b0, (short)0, c0, false, false);
    c1 = __builtin_amdgcn_wmma_f32_16x16x4_f32(false, a, false, b1, (short)0, c1, false, false);
    c2 = __builtin_amdgcn_wmma_f32_16x16x4_f32(false, a, false, b2, (short)0, c2, false, false);
    c3 = __builtin_amdgcn_wmma_f32_16x16x4_f32(false, a, false, b3, (short)0, c3, false, false);
    p += 16 * FDIM;
  }

  // ---- Tail: <=15 rows, uniform loop, plain fp32 adds --------------------
  // Only half==0 lanes store, so folding the tail into a per-lane scalar is
  // safe (half==1 lanes' values are discarded).
  float tacc = 0.0f;
  for (int r = s + bulk; r < e; ++r) {
    tacc += xf[(size_t)r * FDIM];
  }

  const v8f cs = (c0 + c1) + (c2 + c3);

  // C/D layout (documented): VGPR0, lanes 0..15 hold D[0][lane] = colsum.
  const float inv = 1.0f / (float)((cnt < 1) ? 1 : cnt);
  if (half == 0) {
    out[(size_t)g * FDIM + f] = (cs[0] + tacc) * inv;
  }
}

extern "C" void kernel_launch(void* const* d_in, const int* in_sizes, int n_in,
                              void* d_out, int out_size, void* d_ws, size_t ws_size,
                              hipStream_t stream) {
  const float*     x     = (const float*)d_in[0];
  const long long* batch = (const long long*)d_in[1];  // int64 in reference
  float*           out   = (float*)d_out;

  const int n = in_sizes[1];           // number of nodes (len(batch))
  const int G = out_size / FDIM;       // NUM_GRAPHS = 1024

  MeanPoolWmmaKernel<<<G, 512, 0, stream>>>(x, batch, out, n);
}